// GAT_35510789603458
// MI455X (gfx1250) — compile-verified
//
#include <hip/hip_runtime.h>

typedef __attribute__((ext_vector_type(2))) float v2f;
typedef __attribute__((ext_vector_type(8))) float v8f;

#define FEAT 48
#define NEG_SLOPE 0.2f

__device__ __forceinline__ void atomic_max_f32(float* addr, float val) {
  // lowers to global_atomic_max_num_f32 at device scope on gfx1250
  __hip_atomic_fetch_max(addr, val, __ATOMIC_RELAXED, __HIP_MEMORY_SCOPE_AGENT);
}

// ---------------------------------------------------------------------------
// h = X[M x K] @ W[K x 48], one wave per 16x16 output tile, fp32 WMMA.
// A tile layout (V_WMMA_F32_16X16X4_F32): lanes 0-15 -> K pair (0,1),
// lanes 16-31 -> K pair (2,3). B mirrors (K rows striped over lanes).
// D: 8 VGPRs, row = r + 8*(lane>=16), col = lane&15.
// ---------------------------------------------------------------------------
template <int K>
__global__ void gat_gemm_wmma(const float* __restrict__ X,
                              const float* __restrict__ W,
                              float* __restrict__ H, int M) {
  const int wave = (int)((blockIdx.x * blockDim.x + threadIdx.x) >> 5);
  const int lane = threadIdx.x & 31;
  const int mt = wave / (FEAT / 16);
  const int nt = wave % (FEAT / 16);
  const int row0 = mt * 16;
  if (row0 >= M) return;                 // wave-uniform: EXEC stays all-ones
  const int col0 = nt * 16;
  const int half = lane >> 4;            // 0 -> K{0,1}, 1 -> K{2,3}
  const int l16 = lane & 15;

  int arow = row0 + l16;
  if (arow >= M) arow = M - 1;           // safe clamp (M is a multiple of 16 here)
  const float* __restrict__ xrow = X + (long long)arow * K;
  const int col = col0 + l16;

  v8f c = {};
#pragma unroll
  for (int k = 0; k < K; k += 4) {
    v2f a, b;
    a.x = xrow[k + 2 * half + 0];
    a.y = xrow[k + 2 * half + 1];
    b.x = W[(k + 2 * half + 0) * FEAT + col];
    b.y = W[(k + 2 * half + 1) * FEAT + col];
    c = __builtin_amdgcn_wmma_f32_16x16x4_f32(false, a, false, b,
                                              (short)0, c, false, false);
  }
#pragma unroll
  for (int r = 0; r < 8; ++r) {
    int row = row0 + 8 * half + r;
    if (row < M) H[(long long)row * FEAT + col] = c[r];
  }
}

// zero the scatter accumulator, reset per-node max / sum
__global__ void gat_init(float* __restrict__ acc, float* __restrict__ m,
                         float* __restrict__ s, int n) {
  int i = blockIdx.x * blockDim.x + threadIdx.x;
  if (i < n * FEAT) acc[i] = 0.0f;
  if (i < n) { m[i] = -INFINITY; s[i] = 0.0f; }
}

// per-node attention scalars: as[i] = h[i] . a_s ; ad[i] = h[i] . a_d
__global__ void gat_alpha(const float* __restrict__ H,
                          const float* __restrict__ a_s,
                          const float* __restrict__ a_d,
                          float* __restrict__ as_out,
                          float* __restrict__ ad_out, int n) {
  int i = blockIdx.x * blockDim.x + threadIdx.x;
  if (i >= n) return;
  const float* h = H + (long long)i * FEAT;
  float s = 0.0f, d = 0.0f;
#pragma unroll
  for (int f = 0; f < FEAT; ++f) {
    float v = h[f];
    s += v * a_s[f];
    d += v * a_d[f];
  }
  as_out[i] = s;
  ad_out[i] = d;
}

__device__ __forceinline__ void edge_ids(const long long* __restrict__ ei,
                                         int E, int i, int& src, int& dst) {
  if (i < E) { src = (int)ei[i]; dst = (int)ei[E + i]; }
  else       { src = dst = i - E; }     // appended self-loops
}

// pass 1: e = leaky_relu(as[src] + ad[dst]); segment max over dst
__global__ void gat_edge_max(const long long* __restrict__ ei, int E, int n,
                             const float* __restrict__ as,
                             const float* __restrict__ ad,
                             float* __restrict__ ebuf, float* __restrict__ m) {
  int i = blockIdx.x * blockDim.x + threadIdx.x;
  if (i >= E + n) return;
  int src, dst;
  edge_ids(ei, E, i, src, dst);
  float e = as[src] + ad[dst];
  e = (e >= 0.0f) ? e : NEG_SLOPE * e;
  ebuf[i] = e;
  atomic_max_f32(m + dst, e);
}

// pass 2: p = exp(e - m[dst]); segment sum over dst
__global__ void gat_edge_exp(const long long* __restrict__ ei, int E, int n,
                             float* __restrict__ ebuf,
                             const float* __restrict__ m,
                             float* __restrict__ s) {
  int i = blockIdx.x * blockDim.x + threadIdx.x;
  if (i >= E + n) return;
  int src, dst;
  edge_ids(ei, E, i, src, dst);
  float p = __expf(ebuf[i] - m[dst]);
  ebuf[i] = p;
  atomicAdd(s + dst, p);                 // global_atomic_add_f32
}

// pass 3: out[dst] += (p / s[dst]) * h[src]; one thread per (edge, feature)
// so consecutive lanes do coalesced loads of h[src] and coalesced atomics
__global__ void gat_scatter(const long long* __restrict__ ei, int E, int n,
                            const float* __restrict__ ebuf,
                            const float* __restrict__ s,
                            const float* __restrict__ H,
                            float* __restrict__ acc) {
  int t = blockIdx.x * blockDim.x + threadIdx.x;
  int tot = (E + n) * FEAT;
  if (t >= tot) return;
  int i = t / FEAT;
  int f = t - i * FEAT;
  int src, dst;
  edge_ids(ei, E, i, src, dst);
  float alpha = ebuf[i] / fmaxf(s[dst], 1e-16f);
  atomicAdd(acc + (long long)dst * FEAT + f,
            alpha * H[(long long)src * FEAT + f]);
}

// out = elu(acc + bias)
__global__ void gat_finalize(const float* __restrict__ acc,
                             const float* __restrict__ bias,
                             float* __restrict__ out, int n) {
  int t = blockIdx.x * blockDim.x + threadIdx.x;
  if (t >= n * FEAT) return;
  int f = t % FEAT;
  float v = acc[t] + bias[f];
  out[t] = (v > 0.0f) ? v : (__expf(v) - 1.0f);
}

extern "C" void kernel_launch(void* const* d_in, const int* in_sizes, int n_in,
                              void* d_out, int out_size, void* d_ws, size_t ws_size,
                              hipStream_t stream) {
  const float*     x     = (const float*)d_in[0];       // [N,100] f32
  const long long* ei    = (const long long*)d_in[1];   // [2,E]   i64
  const float*     W0    = (const float*)d_in[2];       // [100,48]
  const float*     Wrest = (const float*)d_in[3];       // [4,48,48]
  const float*     a_src = (const float*)d_in[4];       // [5,48]
  const float*     a_dst = (const float*)d_in[5];       // [5,48]
  const float*     bias  = (const float*)d_in[6];       // [5,48]

  const int F0 = 100;
  const int N  = in_sizes[0] / F0;
  const int E  = in_sizes[1] / 2;
  const int TOT = E + N;

  // workspace carve-out (256B aligned slabs)
  char* w = (char*)d_ws;
  auto carve = [&](size_t bytes) {
    void* p = (void*)w;
    w += (bytes + 255) & ~(size_t)255;
    return p;
  };
  float* h    = (float*)carve((size_t)N * FEAT * sizeof(float));
  float* xbuf = (float*)carve((size_t)N * FEAT * sizeof(float));
  float* ebuf = (float*)carve((size_t)TOT * sizeof(float));
  float* mbuf = (float*)carve((size_t)N * sizeof(float));
  float* sbuf = (float*)carve((size_t)N * sizeof(float));
  float* asb  = (float*)carve((size_t)N * sizeof(float));
  float* adb  = (float*)carve((size_t)N * sizeof(float));

  const int B = 256;
  const int mtiles = (N + 15) / 16;
  const int waves = mtiles * (FEAT / 16);
  const int gemm_blocks = (waves * 32 + B - 1) / B;
  const int node_blocks = (N + B - 1) / B;
  const int nf_blocks   = (N * FEAT + B - 1) / B;
  const int edge_blocks = (TOT + B - 1) / B;
  const int scat_blocks = (int)(((long long)TOT * FEAT + B - 1) / B);

  for (int l = 0; l < 5; ++l) {
    const float* xin = (l == 0) ? x : xbuf;
    const float* Wl  = (l == 0) ? W0 : Wrest + (size_t)(l - 1) * FEAT * FEAT;

    if (l == 0)
      gat_gemm_wmma<100><<<gemm_blocks, B, 0, stream>>>(xin, Wl, h, N);
    else
      gat_gemm_wmma<48><<<gemm_blocks, B, 0, stream>>>(xin, Wl, h, N);

    // xin (== xbuf for l>0) is fully consumed by the GEMM; now reuse as acc
    gat_init<<<nf_blocks, B, 0, stream>>>(xbuf, mbuf, sbuf, N);
    gat_alpha<<<node_blocks, B, 0, stream>>>(h, a_src + l * FEAT,
                                             a_dst + l * FEAT, asb, adb, N);
    gat_edge_max<<<edge_blocks, B, 0, stream>>>(ei, E, N, asb, adb, ebuf, mbuf);
    gat_edge_exp<<<edge_blocks, B, 0, stream>>>(ei, E, N, ebuf, mbuf, sbuf);
    gat_scatter<<<scat_blocks, B, 0, stream>>>(ei, E, N, ebuf, sbuf, h, xbuf);

    float* outp = (l == 4) ? (float*)d_out : xbuf;
    gat_finalize<<<nf_blocks, B, 0, stream>>>(xbuf, bias + l * FEAT, outp, N);
  }
}